// VulnHunterOmega_40407052320836
// MI455X (gfx1250) — compile-verified
//
#include <hip/hip_runtime.h>
#include <math.h>

// ---------------------------------------------------------------------------
// VulnHunterOmega for MI455X (gfx1250, wave32)
//
// Phase 1: 512 MiB fp32 reduction (HBM-bound, ~22us at 23.3 TB/s)  -> p[], m[]
// Phase 2: coeff[i,j] = sqrt(p_i p_j) * (m_i, m_j)/hypot(m_i,m_j), i<j
// Phase 3: T   = coeff @ E      (128x128x512, re+im)  via v_wmma_f32_16x16x4_f32
// Phase 4: out = E^T @ T        (512x128x512, re+im)  via v_wmma_f32_16x16x4_f32
//          stored interleaved {re,im} = complex64 layout.
// ---------------------------------------------------------------------------

typedef __attribute__((ext_vector_type(2))) float v2f;
typedef __attribute__((ext_vector_type(8))) float v8f;

#define D_DOM   128
#define L_LEN   1048576
#define S_DIM   512
#define CHUNKS  32          // reduction chunks per domain
#define RED_TPB 256

__device__ inline v8f wmma_f32_k4(v2f a, v2f b, v8f c) {
    // 8 args: (neg_a, A, neg_b, B, c_mod, C, reuse_a, reuse_b)
    return __builtin_amdgcn_wmma_f32_16x16x4_f32(
        false, a, false, b, (short)0, c, false, false);
}

// ---------------------------------------------------------------------------
// Phase 1: per-(domain, chunk) partial sums of v^2 and v.
// 128 domains x 32 chunks = 4096 blocks x 256 threads, float4 (b128) loads.
// ---------------------------------------------------------------------------
__global__ __launch_bounds__(RED_TPB)
void vh_reduce(const float* __restrict__ vulns, float* __restrict__ partials) {
    const int d = blockIdx.y;
    const int c = blockIdx.x;
    const int t = threadIdx.x;
    const int elemsPerChunk = L_LEN / CHUNKS;                 // 32768 floats
    const float4* base = reinterpret_cast<const float4*>(
        vulns + (size_t)d * L_LEN + (size_t)c * elemsPerChunk);
    const int vecPerThread = elemsPerChunk / 4 / RED_TPB;     // 32 x float4

    float s = 0.0f, q = 0.0f;
#pragma unroll 4
    for (int i = 0; i < vecPerThread; ++i) {
        float4 v = base[t + i * RED_TPB];                     // contiguous 4KB/iter
        q = fmaf(v.x, v.x, q);
        q = fmaf(v.y, v.y, q);
        q = fmaf(v.z, v.z, q);
        q = fmaf(v.w, v.w, q);
        s += (v.x + v.y) + (v.z + v.w);
    }

    __shared__ float ls[RED_TPB];
    __shared__ float lq[RED_TPB];
    ls[t] = s; lq[t] = q;
    __syncthreads();
    for (int off = RED_TPB / 2; off > 0; off >>= 1) {
        if (t < off) { ls[t] += ls[t + off]; lq[t] += lq[t + off]; }
        __syncthreads();
    }
    if (t == 0) {
        float2* outp = reinterpret_cast<float2*>(partials);
        outp[d * CHUNKS + c] = make_float2(lq[0], ls[0]);     // (sumsq, sum)
    }
}

// ---------------------------------------------------------------------------
// Phase 1b: deterministic reduce of chunk partials -> p[d], m[d]
// ---------------------------------------------------------------------------
__global__ __launch_bounds__(D_DOM)
void vh_finalize(const float* __restrict__ partials,
                 float* __restrict__ p, float* __restrict__ m) {
    const int d = threadIdx.x;
    const float2* pp = reinterpret_cast<const float2*>(partials);
    float q = 0.0f, s = 0.0f;
#pragma unroll
    for (int c = 0; c < CHUNKS; ++c) {
        float2 v = pp[d * CHUNKS + c];
        q += v.x; s += v.y;
    }
    p[d] = q;
    m[d] = s * (1.0f / (float)L_LEN);
}

// ---------------------------------------------------------------------------
// Phase 2: coeff_{ij} (i<j else 0).
// cos(atan2(mj,mi)) = mi/hyp, sin = mj/hyp; hyp==0 -> theta=0 -> (amp, 0).
// ---------------------------------------------------------------------------
__global__ __launch_bounds__(256)
void vh_coeff(const float* __restrict__ p, const float* __restrict__ m,
              float* __restrict__ cr, float* __restrict__ ci) {
    const int idx = blockIdx.x * 256 + threadIdx.x;           // 16384 entries
    const int i = idx >> 7;
    const int j = idx & 127;
    float crv = 0.0f, civ = 0.0f;
    if (j > i) {
        const float mi = m[i], mj = m[j];
        const float amp = sqrtf(p[i] * p[j]);
        const float hyp = sqrtf(fmaf(mi, mi, mj * mj));
        if (hyp > 0.0f) {
            const float inv = amp / hyp;
            crv = inv * mi;
            civ = inv * mj;
        } else {
            crv = amp;                                        // atan2(0,0)=0
        }
    }
    cr[idx] = crv;
    ci[idx] = civ;
}

// ---------------------------------------------------------------------------
// Phase 3: T = coeff @ E  (M=128, N=512, K=128), real+imag per wave.
// One wave (32 lanes) per 16x16 tile. WMMA fp32, K-step 4, two accumulator
// chains per component to overlap WMMA latency.
// A lane mapping (32-bit): h=lane>>4, r=lane&15; vec elem v -> K = k0+v+2h.
// ---------------------------------------------------------------------------
__global__ __launch_bounds__(32)
void vh_gemm1(const float* __restrict__ E,
              const float* __restrict__ cr, const float* __restrict__ ci,
              float* __restrict__ Tr, float* __restrict__ Ti) {
    const int lane = threadIdx.x;
    const int h = lane >> 4, r = lane & 15;
    const int tm = blockIdx.y * 16;                            // i-rows
    const int tn = blockIdx.x * 16;                            // s-cols

    v8f aR0 = {}, aR1 = {}, aI0 = {}, aI1 = {};
    const int rowA = (tm + r) * D_DOM;
    const int colB = tn + r;

#pragma unroll 2
    for (int k0 = 0; k0 < D_DOM; k0 += 8) {
        const int ka = k0 + 2 * h;        // K block for even chain
        const int kb = ka + 4;            // K block for odd chain
        v2f ar0, ar1, ai0, ai1, b0, b1;
        ar0.x = cr[rowA + ka];      ar0.y = cr[rowA + ka + 1];
        ai0.x = ci[rowA + ka];      ai0.y = ci[rowA + ka + 1];
        ar1.x = cr[rowA + kb];      ar1.y = cr[rowA + kb + 1];
        ai1.x = ci[rowA + kb];      ai1.y = ci[rowA + kb + 1];
        b0.x = E[ka * S_DIM + colB];       b0.y = E[(ka + 1) * S_DIM + colB];
        b1.x = E[kb * S_DIM + colB];       b1.y = E[(kb + 1) * S_DIM + colB];
        aR0 = wmma_f32_k4(ar0, b0, aR0);
        aI0 = wmma_f32_k4(ai0, b0, aI0);
        aR1 = wmma_f32_k4(ar1, b1, aR1);
        aI1 = wmma_f32_k4(ai1, b1, aI1);
    }
    const v8f accR = aR0 + aR1;
    const v8f accI = aI0 + aI1;
#pragma unroll
    for (int v = 0; v < 8; ++v) {
        const int row = tm + v + 8 * h;                        // C/D: M = v + 8h
        Tr[row * S_DIM + tn + r] = accR[v];
        Ti[row * S_DIM + tn + r] = accI[v];
    }
}

// ---------------------------------------------------------------------------
// Phase 4: out = E^T @ T  (M=512, N=512, K=128), complex64 interleaved store.
// A[m][k] = E[k][m]; shared A operand feeds both re and im WMMA chains.
// ---------------------------------------------------------------------------
__global__ __launch_bounds__(32)
void vh_gemm2(const float* __restrict__ E,
              const float* __restrict__ Tr, const float* __restrict__ Ti,
              float* __restrict__ out) {
    const int lane = threadIdx.x;
    const int h = lane >> 4, r = lane & 15;
    const int tm = blockIdx.y * 16;                            // s1
    const int tn = blockIdx.x * 16;                            // s2

    v8f aR0 = {}, aR1 = {}, aI0 = {}, aI1 = {};
    const int colA = tm + r;
    const int colB = tn + r;

#pragma unroll 2
    for (int k0 = 0; k0 < D_DOM; k0 += 8) {
        const int ka = k0 + 2 * h;
        const int kb = ka + 4;
        v2f a0, a1, br0, br1, bi0, bi1;
        a0.x  = E[ka * S_DIM + colA];       a0.y  = E[(ka + 1) * S_DIM + colA];
        a1.x  = E[kb * S_DIM + colA];       a1.y  = E[(kb + 1) * S_DIM + colA];
        br0.x = Tr[ka * S_DIM + colB];      br0.y = Tr[(ka + 1) * S_DIM + colB];
        br1.x = Tr[kb * S_DIM + colB];      br1.y = Tr[(kb + 1) * S_DIM + colB];
        bi0.x = Ti[ka * S_DIM + colB];      bi0.y = Ti[(ka + 1) * S_DIM + colB];
        bi1.x = Ti[kb * S_DIM + colB];      bi1.y = Ti[(kb + 1) * S_DIM + colB];
        aR0 = wmma_f32_k4(a0, br0, aR0);
        aI0 = wmma_f32_k4(a0, bi0, aI0);
        aR1 = wmma_f32_k4(a1, br1, aR1);
        aI1 = wmma_f32_k4(a1, bi1, aI1);
    }
    const v8f accR = aR0 + aR1;
    const v8f accI = aI0 + aI1;
    float2* o = reinterpret_cast<float2*>(out);                // complex64
#pragma unroll
    for (int v = 0; v < 8; ++v) {
        const int row = tm + v + 8 * h;
        o[row * S_DIM + tn + r] = make_float2(accR[v], accI[v]);
    }
}

// ---------------------------------------------------------------------------
extern "C" void kernel_launch(void* const* d_in, const int* in_sizes, int n_in,
                              void* d_out, int out_size, void* d_ws, size_t ws_size,
                              hipStream_t stream) {
    const float* vulns = (const float*)d_in[0];   // [128, 1048576] fp32
    const float* embed = (const float*)d_in[1];   // [128, 512] fp32
    float* ws = (float*)d_ws;

    // workspace layout (float offsets)
    float* partials = ws;               //  8192  (D*CHUNKS*2)
    float* p        = ws + 8192;        //   128
    float* m        = ws + 8320;        //   128
    float* cr       = ws + 8448;        // 16384
    float* ci       = ws + 24832;       // 16384
    float* Tr       = ws + 41216;       // 65536  (128x512)
    float* Ti       = ws + 106752;      // 65536
    // total 172288 floats = ~673 KB

    vh_reduce  <<<dim3(CHUNKS, D_DOM), RED_TPB, 0, stream>>>(vulns, partials);
    vh_finalize<<<1, D_DOM, 0, stream>>>(partials, p, m);
    vh_coeff   <<<(D_DOM * D_DOM) / 256, 256, 0, stream>>>(p, m, cr, ci);
    vh_gemm1   <<<dim3(S_DIM / 16, D_DOM / 16), 32, 0, stream>>>(embed, cr, ci, Tr, Ti);
    vh_gemm2   <<<dim3(S_DIM / 16, S_DIM / 16), 32, 0, stream>>>(embed, Tr, Ti,
                                                                 (float*)d_out);
}